// NeuralStyleLoss_55018531062513
// MI455X (gfx1250) — compile-verified
//
#include <hip/hip_runtime.h>

typedef float v2f __attribute__((ext_vector_type(2)));
typedef float v8f __attribute__((ext_vector_type(8)));

#define HRES 256

// ---------------------------------------------------------------- utilities

__global__ void zero_kernel(float* __restrict__ p, long long n) {
  long long i = (long long)blockIdx.x * blockDim.x + threadIdx.x;
  long long stride = (long long)gridDim.x * blockDim.x;
  for (; i < n; i += stride) p[i] = 0.0f;
}

__device__ __forceinline__ float block_reduce_sum(float v, float* sm) {
  int t = threadIdx.x;
  sm[t] = v; __syncthreads();
  for (int s = blockDim.x >> 1; s > 0; s >>= 1) {
    if (t < s) sm[t] += sm[t + s];
    __syncthreads();
  }
  float r = sm[0];
  __syncthreads();
  return r;
}

__device__ __forceinline__ float block_reduce_min(float v, float* sm) {
  int t = threadIdx.x;
  sm[t] = v; __syncthreads();
  for (int s = blockDim.x >> 1; s > 0; s >>= 1) {
    if (t < s) sm[t] = fminf(sm[t], sm[t + s]);
    __syncthreads();
  }
  float r = sm[0];
  __syncthreads();
  return r;
}

__device__ __forceinline__ float block_reduce_max(float v, float* sm) {
  int t = threadIdx.x;
  sm[t] = v; __syncthreads();
  for (int s = blockDim.x >> 1; s > 0; s >>= 1) {
    if (t < s) sm[t] = fmaxf(sm[t], sm[t + s]);
    __syncthreads();
  }
  float r = sm[0];
  __syncthreads();
  return r;
}

// -------------------------------------------------- Gram via f32 WMMA 16x16x4
// G[n] = F F^T, F = [C, X] row-major. One wave per (triangular 16x16 tile,
// K-chunk); symmetric mirror written at the end (halves WMMA work + reads).
// A lane layout (16x4 f32): lane l -> M = l&15, K pair = (l>>4)*2.
// C/D layout: VGPR p -> M = p + 8*(l>>4), N = l&15.
__global__ __launch_bounds__(32) void gram_wmma_kernel(
    const float* __restrict__ F, float* __restrict__ G,
    int C, long long X, long long kChunk) {
  int tiles = C >> 4;
  // triangular decode: blockIdx.x -> (ti, tj) with ti <= tj (uniform, SALU)
  int p = blockIdx.x;
  int ti = 0;
  int rowlen = tiles;
  while (p >= rowlen) { p -= rowlen; ++ti; --rowlen; }
  int tj = ti + p;

  int n = blockIdx.y;
  long long k0 = (long long)blockIdx.z * kChunk;
  long long k1 = k0 + kChunk; if (k1 > X) k1 = X;

  const float* Fb = F + (long long)n * C * X;
  int lane = threadIdx.x;
  int m  = lane & 15;
  int kp = (lane >> 4) << 1;                      // 0 or 2
  const float* arow = Fb + (long long)(ti * 16 + m) * X;
  const float* brow = Fb + (long long)(tj * 16 + m) * X;

  v8f acc = {};
  long long k = k0;
  for (; k + 32 <= k1; k += 32) {
    // prefetch 8 iterations (one 128B line per row) ahead
    __builtin_prefetch((const void*)(arow + k + 256), 0, 3);
    __builtin_prefetch((const void*)(brow + k + 256), 0, 3);
#pragma unroll
    for (int u = 0; u < 8; ++u) {
      long long kk = k + 4 * u + kp;
      v2f a = { arow[kk], arow[kk + 1] };
      v2f b = { brow[kk], brow[kk + 1] };
      acc = __builtin_amdgcn_wmma_f32_16x16x4_f32(
          false, a, false, b, (short)0, acc, false, false);
    }
  }
  for (; k < k1; k += 4) {                        // tail (unused: X % 32 == 0)
    long long kk = k + kp;
    v2f a = { (kk < k1) ? arow[kk] : 0.0f, (kk + 1 < k1) ? arow[kk + 1] : 0.0f };
    v2f b = { (kk < k1) ? brow[kk] : 0.0f, (kk + 1 < k1) ? brow[kk + 1] : 0.0f };
    acc = __builtin_amdgcn_wmma_f32_16x16x4_f32(
        false, a, false, b, (short)0, acc, false, false);
  }

  int ncol  = lane & 15;
  int mhalf = (lane >> 4) << 3;                   // 0 or 8
  float* Gu = G + ((long long)n * C + ti * 16) * C + tj * 16;   // (ti,tj) block
#pragma unroll
  for (int q = 0; q < 8; ++q)
    atomicAdd(&Gu[(long long)(q + mhalf) * C + ncol], acc[q]);
  if (ti != tj) {
    float* Gl = G + ((long long)n * C + tj * 16) * C + ti * 16; // mirrored block
#pragma unroll
    for (int q = 0; q < 8; ++q)
      atomicAdd(&Gl[(long long)ncol * C + (q + mhalf)], acc[q]);
  }
}

// -------------------------------------------------- style L1 vs target grams
__global__ __launch_bounds__(256) void style_kernel(
    const float* __restrict__ G, const float* __restrict__ Gtgt,
    float* __restrict__ acc, int CC, float invNorm, float w) {
  int n = blockIdx.y;
  const float* g  = G    + (long long)n * CC;
  const float* gt = Gtgt + (long long)n * CC;
  __shared__ float sm[256];
  float s = 0.0f;
  for (int i = blockIdx.x * blockDim.x + threadIdx.x; i < CC;
       i += gridDim.x * blockDim.x)
    s += fabsf(g[i] * invNorm - gt[i]);
  float t = block_reduce_sum(s, sm);
  if (threadIdx.x == 0) atomicAdd(&acc[n], w * t);
}

// -------------------------------------------------- content norm + MSE
__global__ __launch_bounds__(256) void content_kernel(
    const float* __restrict__ a, const float* __restrict__ tgt,
    float* __restrict__ acc, int C, int HW) {
  int c = blockIdx.x, n = blockIdx.y;
  const float* ap = a   + ((long long)n * C + c) * HW;
  const float* tp = tgt + ((long long)n * C + c) * HW;
  __shared__ float sm[256];
  float s = 0.0f, sq = 0.0f;
  for (int i = threadIdx.x; i < HW; i += 256) {
    float x = ap[i]; s += x; sq += x * x;
  }
  float tot  = block_reduce_sum(s,  sm);
  float tot2 = block_reduce_sum(sq, sm);
  float mu  = tot / (float)HW;
  float var = (tot2 - (float)HW * mu * mu) / (float)(HW - 1);
  var = fmaxf(var, 0.0f);
  float inv = 1.0f / sqrtf(sqrtf(var) + 1e-8f);
  float e = 0.0f;
  for (int i = threadIdx.x; i < HW; i += 256) {
    float d = (ap[i] - mu) * inv - tp[i];
    e += d * d;
  }
  float te = block_reduce_sum(e, sm);
  if (threadIdx.x == 0) atomicAdd(&acc[n], te);
}

// -------------------------------------------------- histogram matching loss
__global__ __launch_bounds__(256) void hist_kernel(
    const float* __restrict__ src, const float* __restrict__ tgt,
    float* __restrict__ acc, int HW, int THW) {
  int ch = blockIdx.x;
  const float* s = src + (long long)ch * HW;
  const float* t = tgt + (long long)ch * THW;
  __shared__ float red[256];
  __shared__ int   hs[HRES], ht[HRES];
  __shared__ float sq_[HRES], tq_[HRES], interp[HRES];
  int tid = threadIdx.x;

  float smn = 1e30f, smx = -1e30f, tmn = 1e30f, tmx = -1e30f;
  for (int i = tid; i < HW;  i += 256) { float x = s[i]; smn = fminf(smn, x); smx = fmaxf(smx, x); }
  for (int i = tid; i < THW; i += 256) { float x = t[i]; tmn = fminf(tmn, x); tmx = fmaxf(tmx, x); }
  smn = block_reduce_min(smn, red);
  smx = block_reduce_max(smx, red);
  tmn = block_reduce_min(tmn, red);
  tmx = block_reduce_max(tmx, red);

  hs[tid] = 0; ht[tid] = 0;
  __syncthreads();
  float srange = smx - smn, trange = tmx - tmn;
  for (int i = tid; i < HW; i += 256) {
    int idx = (int)floorf((s[i] - smn) / srange * (float)HRES);
    idx = min(max(idx, 0), HRES - 1);
    atomicAdd(&hs[idx], 1);
  }
  for (int i = tid; i < THW; i += 256) {
    int idx = (int)floorf((t[i] - tmn) / trange * (float)HRES);
    idx = min(max(idx, 0), HRES - 1);
    atomicAdd(&ht[idx], 1);
  }
  __syncthreads();
  if (tid == 0) {
    float cs = 0.0f;
    for (int k = 0; k < HRES; ++k) { cs += (float)hs[k]; sq_[k] = cs; }
    float icn = 1.0f / cs;
    for (int k = 0; k < HRES; ++k) sq_[k] *= icn;
    float ct = 0.0f;
    for (int k = 0; k < HRES; ++k) { ct += (float)ht[k]; tq_[k] = ct; }
    float icT = 1.0f / ct;
    for (int k = 0; k < HRES; ++k) tq_[k] *= icT;
  }
  __syncthreads();
  float tsz = trange / (float)(HRES - 1);
  {
    float sk = sq_[tid];
    int cnt = 0;
    for (int j = 0; j < HRES; ++j) cnt += (tq_[j] < sk) ? 1 : 0;
    interp[tid] = (float)cnt * tsz + tmn;
  }
  __syncthreads();
  float ssz = srange / (float)(HRES - 1);
  float e = 0.0f;
  for (int i = tid; i < HW; i += 256) {
    float x = s[i];
    float binf = (x - smn) / (ssz + 1e-8f);
    int bi = (int)binf; bi = min(max(bi, 0), HRES - 1);
    float low  = interp[bi];
    float diff = interp[min(bi + 1, HRES - 1)] - low;
    float rm = low + (binf - (float)bi) * diff;
    float d = x - rm;
    e += d * d;
  }
  float te = block_reduce_sum(e, red);
  if (tid == 0) atomicAdd(acc, te);
}

// -------------------------------------------------- final scalar combine
__global__ void combine_kernel(const float* __restrict__ style_acc,
                               const float* __restrict__ content_acc,
                               const float* __restrict__ hist_acc,
                               const float* __restrict__ ratio,
                               float* __restrict__ out) {
  float s_ratio = ratio[0];
  float c_ratio = 1.0f - s_ratio;
  float hl = hist_acc[0] * (1.0f / 1048576.0f);          // / (N*C*H*W) = 2*512*1024
  float loss = 0.0f;
  for (int n = 0; n < 2; ++n) {
    float cl = content_acc[n] * (1.0f / 4194304.0f);     // / (256*128*128)
    loss += c_ratio * cl + s_ratio * (style_acc[n] + hl);
  }
  out[0] = loss * 0.5f;
}

// ----------------------------------------------------------------- launcher
extern "C" void kernel_launch(void* const* d_in, const int* in_sizes, int n_in,
                              void* d_out, int out_size, void* d_ws, size_t ws_size,
                              hipStream_t stream) {
  const float* act1   = (const float*)d_in[0];   // [2, 64,512,512]
  const float* act2   = (const float*)d_in[1];   // [2,128,256,256]
  const float* act3   = (const float*)d_in[2];   // [2,256,128,128]
  const float* act3b  = (const float*)d_in[3];   // [2,256,128,128] (content)
  const float* act4   = (const float*)d_in[4];   // [2,512, 64, 64]
  const float* act5   = (const float*)d_in[5];   // [2,512, 32, 32]
  const float* tg1    = (const float*)d_in[6];
  const float* tg2    = (const float*)d_in[7];
  const float* tg3    = (const float*)d_in[8];
  const float* tg4    = (const float*)d_in[9];
  const float* tg5    = (const float*)d_in[10];
  const float* tcont  = (const float*)d_in[11];
  const float* thist  = (const float*)d_in[12];
  const float* ratio  = (const float*)d_in[13];
  float* out = (float*)d_out;

  float* ws = (float*)d_ws;
  long long off = 0;
  float* g1 = ws + off; off += 2LL *  64 *  64;
  float* g2 = ws + off; off += 2LL * 128 * 128;
  float* g3 = ws + off; off += 2LL * 256 * 256;
  float* g4 = ws + off; off += 2LL * 512 * 512;
  float* g5 = ws + off; off += 2LL * 512 * 512;
  float* style_acc   = ws + off; off += 2;
  float* content_acc = ws + off; off += 2;
  float* hist_acc    = ws + off; off += 1;
  long long used = off;

  zero_kernel<<<512, 256, 0, stream>>>(ws, used);

  struct Layer { const float* act; float* g; const float* gt; int C; long long X; };
  Layer layers[5] = {
    { act1, g1, tg1,  64, 512LL * 512 },
    { act2, g2, tg2, 128, 256LL * 256 },
    { act3, g3, tg3, 256, 128LL * 128 },
    { act4, g4, tg4, 512,  64LL *  64 },
    { act5, g5, tg5, 512,  32LL *  32 },
  };

  for (int l = 0; l < 5; ++l) {
    int C = layers[l].C;
    long long X = layers[l].X;
    int tiles = C / 16;
    int triTiles = tiles * (tiles + 1) / 2;
    long long chunks = X / 1024;
    if (chunks < 1) chunks = 1;
    if (chunks > 256) chunks = 256;
    long long kChunk = X / chunks;
    dim3 grid((unsigned)triTiles, 2, (unsigned)chunks);
    gram_wmma_kernel<<<grid, 32, 0, stream>>>(layers[l].act, layers[l].g, C, X, kChunk);
  }

  for (int l = 0; l < 5; ++l) {
    int C = layers[l].C;
    int CC = C * C;
    float invNorm = 1.0f / ((float)C * (float)layers[l].X);
    int gx = (CC + 255) / 256; if (gx > 64) gx = 64;
    dim3 grid((unsigned)gx, 2);
    style_kernel<<<grid, 256, 0, stream>>>(layers[l].g, layers[l].gt,
                                           style_acc, CC, invNorm, 0.2f);
  }

  {
    dim3 grid(256, 2);                    // (C, N) for conv3_2
    content_kernel<<<grid, 256, 0, stream>>>(act3b, tcont, content_acc, 256, 128 * 128);
  }

  hist_kernel<<<1024, 256, 0, stream>>>(act5, thist, hist_acc, 1024, 1024);

  combine_kernel<<<1, 1, 0, stream>>>(style_acc, content_acc, hist_acc, ratio, out);
}